// CreatePairsSum_69389491634769
// MI455X (gfx1250) — compile-verified
//
#include <hip/hip_runtime.h>

typedef float v2f __attribute__((ext_vector_type(2)));
typedef float v8f __attribute__((ext_vector_type(8)));

#define BATCH      16384
#define MAX_JETS   10
#define MAX_PAIRS  45
#define NFEAT      64
#define WAVES_PER_BLOCK 8

// out[b] (45x64) = S[jet_num[b]] (45x10 selection, 0/1) @ inputs[b] (10x64)
// padded to 48x12x64 and computed with V_WMMA_F32_16X16X4_F32 tiles.
__global__ __launch_bounds__(256)
void CreatePairsSum_wmma_kernel(const float* __restrict__ inputs,
                                const int*   __restrict__ dict_vals,
                                const int*   __restrict__ jet_num,
                                float*       __restrict__ out_pairs,
                                float*       __restrict__ out_num)
{
    const int lane = threadIdx.x & 31;
    const int wave = threadIdx.x >> 5;
    const int b    = blockIdx.x * WAVES_PER_BLOCK + wave;   // grid sized exactly

    const int half = lane >> 4;   // 0: lanes 0-15, 1: lanes 16-31
    const int lm   = lane & 15;

    const int n = jet_num[b];                       // 2..10, uniform per wave
    const int* __restrict__ dv = dict_vals + (size_t)(n - 2) * (MAX_PAIRS * 2);

    // ---- Build A fragments: S padded to 48x12, entries {0,1}. ----
    // A 16x4 f32 layout: lane l holds row (l%16), K = (l/16)*2 + r in vgpr r.
    v2f afrag[3][3];
#pragma unroll
    for (int mt = 0; mt < 3; ++mt) {
        const int M = mt * 16 + lm;                 // pair row 0..47
        int d0 = -1, d1 = -1;
        if (M < MAX_PAIRS) { d0 = dv[M * 2]; d1 = dv[M * 2 + 1]; }
#pragma unroll
        for (int kt = 0; kt < 3; ++kt) {
            const int k0 = kt * 4 + half * 2;
            const int k1 = k0 + 1;
            v2f a;
            a[0] = (float)((d0 == k0) + (d1 == k0));
            a[1] = (float)((d0 == k1) + (d1 == k1));
            afrag[mt][kt] = a;
        }
    }

    const float* __restrict__ X = inputs    + (size_t)b * (MAX_JETS  * NFEAT);
    float*       __restrict__ O = out_pairs + (size_t)b * (MAX_PAIRS * NFEAT);

#pragma unroll
    for (int nt = 0; nt < 4; ++nt) {
        // ---- B fragments: X padded to 12x64 (rows 10,11 = 0). ----
        // B 4x16 f32 layout: vgpr r holds row (l/16)*2 + r, col l%16.
        v2f bfrag[3];
#pragma unroll
        for (int kt = 0; kt < 3; ++kt) {
            const int r0  = kt * 4 + half * 2;      // jet row 0..11
            const int col = nt * 16 + lm;
            v2f bb;
            bb[0] = (r0     < MAX_JETS) ? X[(size_t)r0       * NFEAT + col] : 0.0f;
            bb[1] = (r0 + 1 < MAX_JETS) ? X[(size_t)(r0 + 1) * NFEAT + col] : 0.0f;
            bfrag[kt] = bb;
        }

#pragma unroll
        for (int mt = 0; mt < 3; ++mt) {
            v8f c = {};
#pragma unroll
            for (int kt = 0; kt < 3; ++kt) {
                // 8 args: (neg_a, A, neg_b, B, c_mod, C, reuse_a, reuse_b)
                c = __builtin_amdgcn_wmma_f32_16x16x4_f32(
                        false, afrag[mt][kt],
                        false, bfrag[kt],
                        (short)0, c,
                        false, false);
            }
            // C 16x16 f32 layout: vgpr r holds M = r + 8*(l/16), N = l%16.
#pragma unroll
            for (int r = 0; r < 8; ++r) {
                const int M = mt * 16 + half * 8 + r;
                if (M < MAX_PAIRS)
                    O[(size_t)M * NFEAT + nt * 16 + lm] = c[r];
            }
        }
    }

    // pairs_num = C(n,2), one lane per wave
    if (lane == 0)
        out_num[b] = (float)((n * (n - 1)) / 2);
}

extern "C" void kernel_launch(void* const* d_in, const int* in_sizes, int n_in,
                              void* d_out, int out_size, void* d_ws, size_t ws_size,
                              hipStream_t stream) {
    (void)in_sizes; (void)n_in; (void)out_size; (void)d_ws; (void)ws_size;
    const float* inputs    = (const float*)d_in[0];
    const int*   dict_vals = (const int*)  d_in[1];
    const int*   jet_num   = (const int*)  d_in[2];

    float* out_pairs = (float*)d_out;
    float* out_num   = (float*)d_out + (size_t)BATCH * MAX_PAIRS * NFEAT;

    const int blocks = BATCH / WAVES_PER_BLOCK;   // 2048, exact
    CreatePairsSum_wmma_kernel<<<blocks, 256, 0, stream>>>(
        inputs, dict_vals, jet_num, out_pairs, out_num);
}